// ConvAttention_27831388078389
// MI455X (gfx1250) — compile-verified
//
#include <hip/hip_runtime.h>

// CDNA5 WMMA operand types (wave32):
//   A 16x4 f32  -> 2 VGPRs/lane
//   C/D 16x16 f32 -> 8 VGPRs/lane
typedef float v2f __attribute__((ext_vector_type(2)));
typedef float v8f __attribute__((ext_vector_type(8)));

#define NB   32      // batch
#define NC   256     // channels
#define NH   64      // H == W
#define NHW  4096    // 64*64

// ---------------------------------------------------------------------------
// Pass 1: three 1x1 convolutions (dot over C per pixel) -> fx, gx, hx [B,64,64]
// One thread owns 4 consecutive pixels (float4 / b128 loads, fully coalesced).
// Bandwidth-bound: reads x exactly once (134 MB).
// ---------------------------------------------------------------------------
__global__ __launch_bounds__(256) void conv3_kernel(
    const float* __restrict__ x,
    const float* __restrict__ wf, const float* __restrict__ wg,
    const float* __restrict__ wh,
    const float* __restrict__ bf, const float* __restrict__ bg,
    const float* __restrict__ bh,
    float* __restrict__ fx, float* __restrict__ gx, float* __restrict__ hx)
{
    __shared__ float swf[NC], swg[NC], swh[NC];
    const int tid = threadIdx.x;
    swf[tid] = wf[tid];
    swg[tid] = wg[tid];
    swh[tid] = wh[tid];
    __syncthreads();

    const int q  = blockIdx.x * 256 + tid;   // quad index over B*HW/4
    const int p0 = q << 2;                   // first pixel of the quad
    const int b  = p0 >> 12;                 // / 4096 (quads never straddle batches)
    const int hw = p0 & (NHW - 1);
    const float* xp = x + b * (NC * NHW) + hw;

    float4 sf = make_float4(0.f, 0.f, 0.f, 0.f);
    float4 sg = sf, sh = sf;

    #pragma unroll 8
    for (int c = 0; c < NC; ++c) {
        const float4 xv = *(const float4*)(xp + c * NHW);
        const float a = swf[c], bq = swg[c], d = swh[c];
        sf.x = fmaf(xv.x, a, sf.x); sf.y = fmaf(xv.y, a, sf.y);
        sf.z = fmaf(xv.z, a, sf.z); sf.w = fmaf(xv.w, a, sf.w);
        sg.x = fmaf(xv.x, bq, sg.x); sg.y = fmaf(xv.y, bq, sg.y);
        sg.z = fmaf(xv.z, bq, sg.z); sg.w = fmaf(xv.w, bq, sg.w);
        sh.x = fmaf(xv.x, d, sh.x); sh.y = fmaf(xv.y, d, sh.y);
        sh.z = fmaf(xv.z, d, sh.z); sh.w = fmaf(xv.w, d, sh.w);
    }

    const float bfv = bf[0], bgv = bg[0], bhv = bh[0];
    *(float4*)(fx + p0) = make_float4(sf.x + bfv, sf.y + bfv, sf.z + bfv, sf.w + bfv);
    *(float4*)(gx + p0) = make_float4(sg.x + bgv, sg.y + bgv, sg.z + bgv, sg.w + bgv);
    *(float4*)(hx + p0) = make_float4(sh.x + bhv, sh.y + bhv, sh.z + bhv, sh.w + bhv);
}

// ---------------------------------------------------------------------------
// Pass 2: per batch:  S = fx @ gx ; P = softmax_axis1(S) = exp(S - S) ;
//                     O1 = hx @ P     (all 64x64, fp32 WMMA 16x16x4)
// One 512-thread block (16 wave32s) per batch; each wave owns one 16x16 tile.
// LDS double-duty: {fx,gx} then {hx,P} -> 32 KB total.
// ---------------------------------------------------------------------------
__global__ __launch_bounds__(512) void attn_gemm_kernel(
    const float* __restrict__ fx, const float* __restrict__ gx,
    const float* __restrict__ hx, float* __restrict__ o1)
{
    __shared__ float sA[NH * NH];   // fx, later hx
    __shared__ float sB[NH * NH];   // gx, later P
    const int b    = blockIdx.x;
    const int tid  = threadIdx.x;
    const int lane = tid & 31;
    const int wave = tid >> 5;            // 0..15
    const int ti   = (wave >> 2) << 4;    // tile row base
    const int tj   = (wave & 3) << 4;     // tile col base
    const int r    = lane & 15;           // row (A) / col (B) within tile
    const int khi  = (lane >> 4) << 1;    // 0 or 2: K-pair select (ISA A 16x4 layout)

    for (int i = tid; i < NH * NH; i += 512) {
        sA[i] = fx[b * NHW + i];
        sB[i] = gx[b * NHW + i];
    }
    __syncthreads();

    // S tile = fx @ gx
    v8f acc = {0.f, 0.f, 0.f, 0.f, 0.f, 0.f, 0.f, 0.f};
    #pragma unroll
    for (int k = 0; k < NH; k += 4) {
        v2f a, bb;
        a.x  = sA[(ti + r) * NH + k + khi];
        a.y  = sA[(ti + r) * NH + k + khi + 1];
        bb.x = sB[(k + khi) * NH + tj + r];
        bb.y = sB[(k + khi + 1) * NH + tj + r];
        acc = __builtin_amdgcn_wmma_f32_16x16x4_f32(
                  false, a, false, bb, (short)0, acc, false, false);
    }
    __syncthreads();   // all reads of sA/sB done

    // softmax over the singleton channel axis: exp(s - s) (== 1.0 for finite s,
    // propagates NaN like the reference; keeps the first GEMM live).
    #pragma unroll
    for (int v = 0; v < 8; ++v) {
        const float s   = acc[v];
        const int   row = ti + v + ((lane >> 4) << 3);   // M = v + 8*(lane>=16)
        const int   col = tj + (lane & 15);
        sB[row * NH + col] = __expf(s - s);
    }
    for (int i = tid; i < NH * NH; i += 512)
        sA[i] = hx[b * NHW + i];
    __syncthreads();

    // O1 tile = hx @ P
    v8f acc2 = {0.f, 0.f, 0.f, 0.f, 0.f, 0.f, 0.f, 0.f};
    #pragma unroll
    for (int k = 0; k < NH; k += 4) {
        v2f a, bb;
        a.x  = sA[(ti + r) * NH + k + khi];
        a.y  = sA[(ti + r) * NH + k + khi + 1];
        bb.x = sB[(k + khi) * NH + tj + r];
        bb.y = sB[(k + khi + 1) * NH + tj + r];
        acc2 = __builtin_amdgcn_wmma_f32_16x16x4_f32(
                   false, a, false, bb, (short)0, acc2, false, false);
    }

    #pragma unroll
    for (int v = 0; v < 8; ++v) {
        const int row = ti + v + ((lane >> 4) << 3);
        const int col = tj + (lane & 15);
        o1[b * NHW + row * NH + col] = acc2[v];
    }
}

// ---------------------------------------------------------------------------
// Pass 3: out[b,c,h,w] = O1[b,h,w] * x[b,c,h,w]   (float4, fully coalesced)
// ---------------------------------------------------------------------------
__global__ __launch_bounds__(256) void scale_kernel(
    const float* __restrict__ x, const float* __restrict__ o1,
    float* __restrict__ out)
{
    const int i  = (blockIdx.x * 256 + threadIdx.x) << 2;  // element index
    const int hw = i & (NHW - 1);
    const int bc = i >> 12;          // b*C + c
    const int b  = bc >> 8;          // / 256
    const float4 xv = *(const float4*)(x + i);
    const float4 sv = *(const float4*)(o1 + b * NHW + hw);
    *(float4*)(out + i) = make_float4(xv.x * sv.x, xv.y * sv.y,
                                      xv.z * sv.z, xv.w * sv.w);
}

// ---------------------------------------------------------------------------
extern "C" void kernel_launch(void* const* d_in, const int* in_sizes, int n_in,
                              void* d_out, int out_size, void* d_ws, size_t ws_size,
                              hipStream_t stream) {
    const float* x  = (const float*)d_in[0];
    const float* wf = (const float*)d_in[1];
    const float* bf = (const float*)d_in[2];
    const float* wg = (const float*)d_in[3];
    const float* bg = (const float*)d_in[4];
    const float* wh = (const float*)d_in[5];
    const float* bh = (const float*)d_in[6];
    float* out = (float*)d_out;

    // Workspace layout (floats): fx | gx | hx | o1, each B*HW = 131072 -> 2 MB.
    float* fx = (float*)d_ws;
    float* gx = fx + NB * NHW;
    float* hx = gx + NB * NHW;
    float* o1 = hx + NB * NHW;

    // Pass 1: B*HW/4 = 32768 quad-threads -> 128 blocks of 256.
    conv3_kernel<<<(NB * NHW / 4) / 256, 256, 0, stream>>>(
        x, wf, wg, wh, bf, bg, bh, fx, gx, hx);

    // Pass 2: one block per batch, 16 wave32s.
    attn_gemm_kernel<<<NB, 512, 0, stream>>>(fx, gx, hx, o1);

    // Pass 3: B*C*HW/4 = 8388608 quad-threads -> 32768 blocks of 256.
    scale_kernel<<<(NB * NC * NHW / 4) / 256, 256, 0, stream>>>(x, o1, out);
}